// MTAD_GAT_2439541424426
// MI455X (gfx1250) — compile-verified
//
#include <hip/hip_runtime.h>
#include <math.h>

#define FEATS 64
#define NNODE 65
#define HSZ   4096
#define ISZ   12480   // (F+1)*F*3
#define TH3   12288   // 3*H

typedef float v2f __attribute__((ext_vector_type(2)));
typedef float v4f __attribute__((ext_vector_type(4)));
typedef float v8f __attribute__((ext_vector_type(8)));

__device__ __forceinline__ float lrelu(float v) { return v > 0.f ? v : 0.2f * v; }

// ---------------------------------------------------------------------------
// Kernel 1: both GATs via V_WMMA_F32_16X16X4_F32 + column softmax; builds x.
// One block, 512 threads = 16 waves; wave w owns 16x16 output tile (w/4, w%4).
// ---------------------------------------------------------------------------
__global__ __launch_bounds__(512)
void gat_build_x(const float* __restrict__ data,
                 const float* __restrict__ fW, const float* __restrict__ fal,
                 const float* __restrict__ far_, const float* __restrict__ fb,
                 const float* __restrict__ tW, const float* __restrict__ tal,
                 const float* __restrict__ tar_, const float* __restrict__ tb,
                 float* __restrict__ x)
{
    __shared__ float hS[FEATS][FEATS + 1];   // +1 pad: column reads hit 64 banks
    const int tid  = threadIdx.x;
    const int lane = tid & 31;
    const int wave = tid >> 5;               // 0..15

    // channel 0 of x: data_r (row 0 zeros, rows 1..64 = data)
    for (int idx = tid; idx < NNODE * FEATS; idx += 512) {
        int n = idx >> 6, f = idx & 63;
        x[idx * 3 + 0] = (n == 0) ? 0.f : data[(n - 1) * FEATS + f];
    }

    const int tm   = wave >> 2, tn = wave & 3;
    const int mrow = tm * 16 + (lane & 15);      // A: M index for this lane
    const int ncol = tn * 16 + (lane & 15);      // B: N index for this lane
    const int kk   = (lane >> 4) * 2;            // lanes 16-31 carry K+2,K+3

    for (int br = 0; br < 2; ++br) {             // 0 = feature GAT, 1 = time GAT
        const float* W  = br ? tW  : fW;
        const float* al = br ? tal : fal;
        const float* b  = br ? tb  : fb;

        // h1 = (br ? data^T : data) @ W   (fp32 WMMA, K in steps of 4)
        v8f c = {};
        for (int k0 = 0; k0 < FEATS; k0 += 4) {
            v2f a, bf;
            if (br == 0) {                       // uniform branch: EXEC stays all-1s
                a.x = data[mrow * FEATS + k0 + kk];
                a.y = data[mrow * FEATS + k0 + kk + 1];
            } else {                             // A = data^T
                a.x = data[(k0 + kk)     * FEATS + mrow];
                a.y = data[(k0 + kk + 1) * FEATS + mrow];
            }
            bf.x = W[(k0 + kk)     * FEATS + ncol];
            bf.y = W[(k0 + kk + 1) * FEATS + ncol];
            c = __builtin_amdgcn_wmma_f32_16x16x4_f32(false, a, false, bf,
                                                      (short)0, c, false, false);
        }
        __syncthreads();                         // hS free from previous branch
        #pragma unroll
        for (int r = 0; r < 8; ++r) {            // D layout: VGPR r -> M = r (+8 hi lanes)
            int row = tm * 16 + r + ((lane >= 16) ? 8 : 0);
            int col = tn * 16 + (lane & 15);
            hS[row][col] = c[r];
        }
        __syncthreads();

        const int ch = 1 + br;
        // node 0 output: softmax over 65 logits per column.
        // h[0] = 0 -> er[0] = 0, so e[n] = lrelu(h[n]*al), and e[0] = 0.
        if (tid < FEATS) {
            float alf = al[tid];
            float m = 0.f;                                  // node-0 logit
            for (int i = 0; i < FEATS; ++i)
                m = fmaxf(m, lrelu(hS[i][tid] * alf));
            float s   = __expf(0.f - m);                    // node-0 term (h=0)
            float num = 0.f;
            for (int i = 0; i < FEATS; ++i) {
                float e = __expf(lrelu(hS[i][tid] * alf) - m);
                s   += e;
                num += e * hS[i][tid];
            }
            x[tid * 3 + ch] = num / s + b[tid];             // node 0, feature tid
        }
        // nodes 1..64: self-loop only -> out = h + b
        for (int idx = tid; idx < FEATS * FEATS; idx += 512) {
            int i = idx >> 6, f = idx & 63;
            x[((1 + i) * FEATS + f) * 3 + ch] = hS[i][f] + b[f];
        }
        __syncthreads();
    }
}

// ---------------------------------------------------------------------------
// Kernel 2: the bandwidth-bound GEMVs. One wave per output row.
// Rows [0,12288): gi = W_ih @ x + b_ih   (K = 12480)
// Rows [12288,24576): gh = W_hh @ h0 + b_hh (K = 4096)
// Weight stream uses non-temporal loads (zero reuse); x/h0 stay L2-resident.
// ---------------------------------------------------------------------------
__global__ __launch_bounds__(256)
void gru_gemv(const float* __restrict__ W_ih, const float* __restrict__ W_hh,
              const float* __restrict__ b_ih, const float* __restrict__ b_hh,
              const float* __restrict__ x,    const float* __restrict__ h0,
              float* __restrict__ gi, float* __restrict__ gh)
{
    const int lane = threadIdx.x & 31;
    const int row  = blockIdx.x * 8 + (threadIdx.x >> 5);

    const float* Wr; const float* vec; float* out;
    int K, oi;
    if (row < TH3) {
        Wr  = W_ih + (size_t)row * ISZ;  vec = x;  K = ISZ;
        out = gi;  oi = row;
    } else {
        int r2 = row - TH3;
        Wr  = W_hh + (size_t)r2 * HSZ;   vec = h0; K = HSZ;
        out = gh;  oi = r2;
    }
    float bias = (row < TH3) ? b_ih[oi] : b_hh[oi];

    float acc = 0.f;
    for (int k = lane * 4; k + 4 <= K; k += 128) {          // 512B/wave/step, coalesced
        v4f w = __builtin_nontemporal_load((const v4f*)(Wr + k));
        v4f v = *(const v4f*)(vec + k);
        acc += w.x * v.x + w.y * v.y + w.z * v.z + w.w * v.w;
    }
    #pragma unroll
    for (int off = 16; off > 0; off >>= 1)
        acc += __shfl_down(acc, off, 32);
    if (lane == 0) out[oi] = acc + bias;
}

// ---------------------------------------------------------------------------
// Kernel 3: fused GRU gate math; writes h_new twice (tuple output).
// ---------------------------------------------------------------------------
__global__ __launch_bounds__(256)
void gru_combine(const float* __restrict__ gi, const float* __restrict__ gh,
                 const float* __restrict__ h0, float* __restrict__ out)
{
    int i = blockIdx.x * blockDim.x + threadIdx.x;
    if (i >= HSZ) return;
    float r  = 1.f / (1.f + __expf(-(gi[i]           + gh[i])));
    float z  = 1.f / (1.f + __expf(-(gi[HSZ + i]     + gh[HSZ + i])));
    float n  = tanhf(gi[2 * HSZ + i] + r * gh[2 * HSZ + i]);
    float hn = (1.f - z) * n + z * h0[i];
    out[i]        = hn;
    out[HSZ + i]  = hn;
}

extern "C" void kernel_launch(void* const* d_in, const int* in_sizes, int n_in,
                              void* d_out, int out_size, void* d_ws, size_t ws_size,
                              hipStream_t stream) {
    const float* data   = (const float*)d_in[0];
    const float* hidden = (const float*)d_in[1];
    const float* fW     = (const float*)d_in[2];
    const float* fal    = (const float*)d_in[3];
    const float* far_   = (const float*)d_in[4];
    const float* fb     = (const float*)d_in[5];
    const float* tW     = (const float*)d_in[6];
    const float* tal    = (const float*)d_in[7];
    const float* tar_   = (const float*)d_in[8];
    const float* tb     = (const float*)d_in[9];
    const float* W_ih   = (const float*)d_in[10];
    const float* W_hh   = (const float*)d_in[11];
    const float* b_ih   = (const float*)d_in[12];
    const float* b_hh   = (const float*)d_in[13];

    float* ws = (float*)d_ws;
    float* x  = ws;                    // 12480 floats
    float* gi = ws + ISZ;              // 12288 floats
    float* gh = ws + ISZ + TH3;        // 12288 floats

    gat_build_x<<<1, 512, 0, stream>>>(data, fW, fal, far_, fb,
                                       tW, tal, tar_, tb, x);
    gru_gemv<<<(2 * TH3) / 8, 256, 0, stream>>>(W_ih, W_hh, b_ih, b_hh,
                                                x, hidden, gi, gh);
    gru_combine<<<(HSZ + 255) / 256, 256, 0, stream>>>(gi, gh, hidden,
                                                       (float*)d_out);
}